// AGCNConv_48610439856570
// MI455X (gfx1250) — compile-verified
//
#include <hip/hip_runtime.h>
#include <hip/hip_bf16.h>

typedef float v2f __attribute__((ext_vector_type(2)));
typedef float v8f __attribute__((ext_vector_type(8)));

#define NNODES 40000
#define NEDGES 640000
#define DFEAT  128
#define LN_EPS 1e-5f
#define LEAKY  0.2f

// ---------------------------------------------------------------- zero init
__global__ void zero_f32_kernel(float* __restrict__ p, int n) {
    int i = blockIdx.x * blockDim.x + threadIdx.x;
    if (i < n) p[i] = 0.0f;
}

// ---------------------------------------------------------------- GEMM: h = x @ W  (fp32 WMMA 16x16x4)
// grid: 2500 blocks (16-row tiles), 256 threads = 8 waves, wave w owns cols [16w,16w+16)
__global__ void gemm_wmma_kernel(const float* __restrict__ x,
                                 const float* __restrict__ W,
                                 float* __restrict__ h) {
    const int row0 = blockIdx.x * 16;
    const int wave = threadIdx.x >> 5;      // 0..7 -> col tile
    const int lane = threadIdx.x & 31;
    const int col0 = wave * 16;
    const int l    = lane & 15;             // M index (A) / N index (B,D)
    const int kh   = lane >> 4;             // 0: K pair {0,1}, 1: K pair {2,3}

    v8f acc = {};
    const float* xrow = x + (size_t)(row0 + l) * DFEAT;

    #pragma unroll 8
    for (int k = 0; k < DFEAT; k += 4) {
        // A fragment 16x4: v0 = A[M=l][k+2kh], v1 = A[M=l][k+2kh+1]
        v2f a;
        a.x = xrow[k + 2 * kh];
        a.y = xrow[k + 2 * kh + 1];
        // B fragment 4x16: v0 = B[K=k+2kh][col0+l], v1 = B[K=k+2kh+1][col0+l]
        v2f b;
        b.x = W[(size_t)(k + 2 * kh) * DFEAT + col0 + l];
        b.y = W[(size_t)(k + 2 * kh + 1) * DFEAT + col0 + l];
        acc = __builtin_amdgcn_wmma_f32_16x16x4_f32(
            /*neg_a=*/false, a, /*neg_b=*/false, b,
            /*c_mod=*/(short)0, acc, /*reuse_a=*/false, /*reuse_b=*/false);
    }

    // D 16x16: lanes 0-15 -> M=r, lanes 16-31 -> M=8+r, N = col0+l
    #pragma unroll
    for (int r = 0; r < 8; ++r) {
        h[(size_t)(row0 + kh * 8 + r) * DFEAT + col0 + l] = acc[r];
    }
}

// ---------------------------------------------------------------- degree scatter
__global__ void degree_kernel(const int* __restrict__ ei,
                              const float* __restrict__ ew,
                              const float* __restrict__ aw,
                              float* __restrict__ deg, int E) {
    int e = blockIdx.x * blockDim.x + threadIdx.x;
    if (e >= E) return;
    float s = 1.0f / (1.0f + __expf(-aw[0]));   // sigmoid(adaptive_weight)
    unsafeAtomicAdd(&deg[ei[e]], ew[e] * s);    // row = ei[0:E]
}

// ---------------------------------------------------------------- deg -> deg^{-1/2} in place
__global__ void dis_kernel(float* __restrict__ deg, int n) {
    int i = blockIdx.x * blockDim.x + threadIdx.x;
    if (i >= n) return;
    float d = deg[i];
    deg[i] = (d > 0.0f) ? rsqrtf(d) : 0.0f;
}

// ---------------------------------------------------------------- propagate: acc[row] += h[col] * norm
// one wave per edge, 4 features per lane
__global__ void propagate_kernel(const int* __restrict__ ei,
                                 const float* __restrict__ ew,
                                 const float* __restrict__ aw,
                                 const float* __restrict__ dis,
                                 const float* __restrict__ h,
                                 float* __restrict__ acc, int E) {
    int e = blockIdx.x * 8 + (threadIdx.x >> 5);
    int lane = threadIdx.x & 31;
    if (e >= E) return;
    int r = ei[e];
    int c = ei[E + e];
    float s = 1.0f / (1.0f + __expf(-aw[0]));
    float norm = dis[r] * (ew[e] * s) * dis[c];

    const float4 hv = reinterpret_cast<const float4*>(h + (size_t)c * DFEAT)[lane];
    float* dst = acc + (size_t)r * DFEAT + lane * 4;
    unsafeAtomicAdd(dst + 0, hv.x * norm);
    unsafeAtomicAdd(dst + 1, hv.y * norm);
    unsafeAtomicAdd(dst + 2, hv.z * norm);
    unsafeAtomicAdd(dst + 3, hv.w * norm);
}

// ---------------------------------------------------------------- bias + LayerNorm + LeakyReLU
// one wave per node, 4 features per lane, wave32 shuffle reductions
__global__ void bias_ln_kernel(const float* __restrict__ acc,
                               const float* __restrict__ bias,
                               const float* __restrict__ gamma,
                               const float* __restrict__ beta,
                               float* __restrict__ out, int n) {
    int node = blockIdx.x * 8 + (threadIdx.x >> 5);
    int lane = threadIdx.x & 31;
    if (node >= n) return;

    float4 v = reinterpret_cast<const float4*>(acc + (size_t)node * DFEAT)[lane];
    float4 b = reinterpret_cast<const float4*>(bias)[lane];
    v.x += b.x; v.y += b.y; v.z += b.z; v.w += b.w;

    float sum = v.x + v.y + v.z + v.w;
    #pragma unroll
    for (int off = 16; off > 0; off >>= 1) sum += __shfl_xor(sum, off, 32);
    float mu = sum * (1.0f / DFEAT);

    float dx = v.x - mu, dy = v.y - mu, dz = v.z - mu, dw = v.w - mu;
    float sq = dx * dx + dy * dy + dz * dz + dw * dw;
    #pragma unroll
    for (int off = 16; off > 0; off >>= 1) sq += __shfl_xor(sq, off, 32);
    float inv = rsqrtf(sq * (1.0f / DFEAT) + LN_EPS);

    float4 g = reinterpret_cast<const float4*>(gamma)[lane];
    float4 bt = reinterpret_cast<const float4*>(beta)[lane];
    float4 o;
    o.x = dx * inv * g.x + bt.x;
    o.y = dy * inv * g.y + bt.y;
    o.z = dz * inv * g.z + bt.z;
    o.w = dw * inv * g.w + bt.w;
    o.x = (o.x >= 0.0f) ? o.x : LEAKY * o.x;
    o.y = (o.y >= 0.0f) ? o.y : LEAKY * o.y;
    o.z = (o.z >= 0.0f) ? o.z : LEAKY * o.z;
    o.w = (o.w >= 0.0f) ? o.w : LEAKY * o.w;
    reinterpret_cast<float4*>(out + (size_t)node * DFEAT)[lane] = o;
}

extern "C" void kernel_launch(void* const* d_in, const int* in_sizes, int n_in,
                              void* d_out, int out_size, void* d_ws, size_t ws_size,
                              hipStream_t stream) {
    const float* x    = (const float*)d_in[0];
    const int*   ei   = (const int*)d_in[1];    // [2, E]: rows then cols
    const float* ew   = (const float*)d_in[2];
    const float* W    = (const float*)d_in[3];
    const float* aw   = (const float*)d_in[4];
    const float* bias = (const float*)d_in[5];
    const float* lng  = (const float*)d_in[6];
    const float* lnb  = (const float*)d_in[7];
    float* out = (float*)d_out;

    // workspace: acc [N*D] floats, then deg/dis [N] floats
    float* acc = (float*)d_ws;
    float* deg = acc + (size_t)NNODES * DFEAT;
    // h is staged in d_out (fully consumed by propagate before bias_ln overwrites it)
    float* h = out;

    const int zero_n = NNODES * DFEAT + NNODES;
    zero_f32_kernel<<<(zero_n + 255) / 256, 256, 0, stream>>>(acc, zero_n);

    gemm_wmma_kernel<<<NNODES / 16, 256, 0, stream>>>(x, W, h);

    degree_kernel<<<(NEDGES + 255) / 256, 256, 0, stream>>>(ei, ew, aw, deg, NEDGES);

    dis_kernel<<<(NNODES + 255) / 256, 256, 0, stream>>>(deg, NNODES);

    propagate_kernel<<<NEDGES / 8, 256, 0, stream>>>(ei, ew, aw, deg, h, acc, NEDGES);

    bias_ln_kernel<<<NNODES / 8, 256, 0, stream>>>(acc, bias, lng, lnb, out, NNODES);
}